// GeoCyclicPadding_4982162063571
// MI455X (gfx1250) — compile-verified
//
#include <hip/hip_runtime.h>

// GeoCyclicPadding: x (2,192,360,720) f32, p=3 -> out (2,192,366,726) f32.
// Pure gather, 0 FLOPs, ~806 MB traffic -> HBM-bound (~35us floor @ 23.3 TB/s).
// Strategy: one workgroup per output row. Stage the (single) source input row
// (720 f32 = 2880 B, 16B-aligned) into LDS via gfx1250 async global->LDS DMA
// (ASYNCcnt + s_wait_asynccnt), then write the width-permuted 726-wide output
// row with coalesced non-temporal b32 stores. LDS absorbs the wrap seams.

typedef __attribute__((ext_vector_type(4))) float v4f;
typedef __attribute__((ext_vector_type(4))) int   v4i;
// hipcc prints AS1 as "__device__"; builtin wants non-const int4 pointers.
typedef __attribute__((address_space(1))) v4i* glb_v4i_ptr;
typedef __attribute__((address_space(3))) v4i* lds_v4i_ptr;

#if __has_builtin(__builtin_amdgcn_global_load_async_to_lds_b128)
#define GEO_ASYNC_LDS 1
#else
#define GEO_ASYNC_LDS 0
#endif

namespace geo {
constexpr int B  = 2;
constexpr int C  = 192;
constexpr int H  = 360;
constexpr int W  = 720;          // row = 2880 B, multiple of 16 -> b128 OK
constexpr int P  = 3;
constexpr int HO = H + 2 * P;    // 366
constexpr int WO = W + 2 * P;    // 726
constexpr int MID = W / 2;       // 360
}  // namespace geo

__global__ __launch_bounds__(256) void geo_cyclic_pad_kernel(
    const float* __restrict__ x, float* __restrict__ out) {
  using namespace geo;
  __shared__ alignas(16) float row[W];

  const int ho  = blockIdx.x;   // 0 .. 365
  const int bc  = blockIdx.y;   // 0 .. 383 (b*C + c)
  const int tid = threadIdx.x;

  const bool interior = (ho >= P) && (ho < P + H);
  int src_h;
  if (interior)     src_h = ho - P;               // cp body
  else if (ho < P)  src_h = P - 1 - ho;           // top rows: (p-i-1)%H = {2,1,0}
  else              src_h = H - 1 - (ho - P - H); // bottom: (H-i-1)%H = {359,358,357}

  const float* src_row = x + ((size_t)bc * H + src_h) * W;

  // ---- stage source row into LDS (180 x b128 = 2880 B) ----
  if (tid < W / 4) {
#if GEO_ASYNC_LDS
    // gfx1250 async DMA: global -> LDS, tracked by ASYNCcnt.
    glb_v4i_ptr gsrc =
        (glb_v4i_ptr)(unsigned long long)((const v4f*)src_row + tid);
    lds_v4i_ptr ldst =
        (lds_v4i_ptr)(unsigned)(unsigned long long)((v4f*)row + tid);
    __builtin_amdgcn_global_load_async_to_lds_b128(gsrc, ldst, 0, 0);
#else
    ((v4f*)row)[tid] = __builtin_nontemporal_load((const v4f*)src_row + tid);
#endif
  }
#if GEO_ASYNC_LDS
#if __has_builtin(__builtin_amdgcn_s_wait_asynccnt)
  __builtin_amdgcn_s_wait_asynccnt(0);
#else
  asm volatile("s_wait_asynccnt 0x0" ::: "memory");
#endif
#endif
  __syncthreads();

  // ---- emit permuted 726-wide output row, coalesced NT stores ----
  float* out_row = out + ((size_t)bc * HO + ho) * WO;
  for (int wo = tid; wo < WO; wo += 256) {
    int sw;
    if (interior) {
      sw = wo - P;                       // cp: (wo-p) mod W
    } else {
      int t = wo + MID;                  // rolled: ((wo+mid) mod (W+2p) - p) mod W
      if (t >= WO) t -= WO;
      sw = t - P;
    }
    if (sw < 0) sw += W;
    else if (sw >= W) sw -= W;           // sw in [-3, 723) -> single fixup suffices
    __builtin_nontemporal_store(row[sw], &out_row[wo]);
  }
}

extern "C" void kernel_launch(void* const* d_in, const int* in_sizes, int n_in,
                              void* d_out, int out_size, void* d_ws, size_t ws_size,
                              hipStream_t stream) {
  (void)in_sizes; (void)n_in; (void)d_ws; (void)ws_size; (void)out_size;
  const float* x = (const float*)d_in[0];  // d_in[1] = pad_width (=3), baked in
  float* out = (float*)d_out;
  dim3 grid(geo::HO, geo::B * geo::C, 1);  // 366 x 384 row-blocks
  geo_cyclic_pad_kernel<<<grid, 256, 0, stream>>>(x, out);
}